// acLSTM_11862699671611
// MI455X (gfx1250) — compile-verified
//
#include <hip/hip_runtime.h>
#include <stdint.h>

#define B_   32
#define T0_  50
#define F_   171
#define FP_  176
#define H_   1024

typedef __attribute__((ext_vector_type(2))) float v2f;
typedef __attribute__((ext_vector_type(8))) float v8f;

// ---------------- workspace layout (float offsets) ----------------
#define OFF_WIH1P 0
#define SZ_WIH1P  (4096*FP_)                 // padded Wih1 [4096][176]
#define OFF_WDP   (OFF_WIH1P + SZ_WIH1P)
#define SZ_WDP    (FP_*H_)                   // padded Wd   [176][1024]
#define OFF_BIAS  (OFF_WDP + SZ_WDP)
#define SZ_BIAS   (3*4096)                   // bih+bhh per layer
#define OFF_BD    (OFF_BIAS + SZ_BIAS)
#define SZ_BD     FP_
#define OFF_HT    (OFF_BD + SZ_BD)
#define SZ_HT     (3*2*32768)                // h transposed, k-pair interleaved, 2 parities
#define OFF_C     (OFF_HT + SZ_HT)
#define SZ_C      (3*32768)                  // c row-major [32][1024] per layer
#define OFF_XT    (OFF_C + SZ_C)
#define SZ_XT     5632                       // x_t transposed padded [176][32] (pair layout)
#define OFF_BAR   (OFF_XT + SZ_XT)           // 2 x u32 barrier state (8B aligned)

__device__ __forceinline__ float sigm_(float x) { return 1.0f / (1.0f + __expf(-x)); }
__device__ __forceinline__ float tanh_(float x) { float e = __expf(2.0f * x); return 1.0f - 2.0f / (e + 1.0f); }

// k-pair interleaved transposed index: element (k, m) of a [K][32] matrix
__device__ __forceinline__ int htix(int k, int m) { return ((k >> 1) << 6) + (m << 1) + (k & 1); }

__device__ __forceinline__ v8f wmma4(v2f a, v2f b, v8f c) {
    return __builtin_amdgcn_wmma_f32_16x16x4_f32(false, a, false, b, (short)0, c, false, false);
}

// grid-wide sense-reversing barrier
__device__ __forceinline__ void gbar(unsigned* bar, unsigned nb) {
    __threadfence();
    __syncthreads();
    if (threadIdx.x == 0) {
        unsigned g   = __hip_atomic_load(&bar[1], __ATOMIC_RELAXED, __HIP_MEMORY_SCOPE_AGENT);
        unsigned old = __hip_atomic_fetch_add(&bar[0], 1u, __ATOMIC_ACQ_REL, __HIP_MEMORY_SCOPE_AGENT);
        if (old == nb - 1u) {
            __hip_atomic_store(&bar[0], 0u, __ATOMIC_RELAXED, __HIP_MEMORY_SCOPE_AGENT);
            __hip_atomic_fetch_add(&bar[1], 1u, __ATOMIC_ACQ_REL, __HIP_MEMORY_SCOPE_AGENT);
        } else {
            while (__hip_atomic_load(&bar[1], __ATOMIC_RELAXED, __HIP_MEMORY_SCOPE_AGENT) == g) {
                __builtin_amdgcn_s_sleep(1);
            }
        }
    }
    __syncthreads();
    __threadfence();
}

// 4-gate fp32 WMMA K-loop (compile-time K and leading dim so the compiler
// emits scalar-counted, unrolled, software-pipelined loops for all layers):
// acc[g] += A(16x4 of S) x B(4x16 of W rows g*H+j0..)
template<int K, int LD>
__device__ __forceinline__ void mm4(const float* __restrict__ S, const float* __restrict__ W,
                                    int m0, int j0, int nl, int hi,
                                    v8f& ai, v8f& af, v8f& ag, v8f& ao) {
    const float* sp = S + hi * 64 + (m0 + nl) * 2;
    const float* wi = W + (size_t)(0 * H_ + j0 + nl) * LD + 2 * hi;
    const float* wf = W + (size_t)(1 * H_ + j0 + nl) * LD + 2 * hi;
    const float* wg = W + (size_t)(2 * H_ + j0 + nl) * LD + 2 * hi;
    const float* wo = W + (size_t)(3 * H_ + j0 + nl) * LD + 2 * hi;
#pragma unroll 4
    for (int kk = 0; kk < K; kk += 4) {
        v2f a  = *(const v2f*)sp;        sp += 128;
        v2f b0 = *(const v2f*)(wi + kk);
        v2f b1 = *(const v2f*)(wf + kk);
        v2f b2 = *(const v2f*)(wg + kk);
        v2f b3 = *(const v2f*)(wo + kk);
        ai = wmma4(a, b0, ai);
        af = wmma4(a, b1, af);
        ag = wmma4(a, b2, ag);
        ao = wmma4(a, b3, ao);
    }
}

// ---------------- prepack: pad weights, combine biases, reset state ----------------
__global__ void acLSTM_prepack(const float* __restrict__ iseq, const int* __restrict__ Gp,
                               const float* __restrict__ Wih1,
                               const float* __restrict__ bih1, const float* __restrict__ bhh1,
                               const float* __restrict__ bih2, const float* __restrict__ bhh2,
                               const float* __restrict__ bih3, const float* __restrict__ bhh3,
                               const float* __restrict__ Wd,   const float* __restrict__ bd,
                               float* __restrict__ out, float* __restrict__ ws) {
    int tid = blockIdx.x * blockDim.x + threadIdx.x;
    int np  = gridDim.x * blockDim.x;
    if (tid == 0) {
        unsigned* bar = (unsigned*)(ws + OFF_BAR);
        bar[0] = 0u; bar[1] = 0u;
    }
    for (int i = tid; i < 4096 * FP_; i += np) {           // padded Wih1
        int n = i / FP_, k = i - n * FP_;
        ws[OFF_WIH1P + i] = (k < F_) ? Wih1[n * F_ + k] : 0.0f;
    }
    for (int i = tid; i < FP_ * H_; i += np) {             // padded Wd
        int n = i >> 10;
        ws[OFF_WDP + i] = (n < F_) ? Wd[i] : 0.0f;
    }
    for (int i = tid; i < 4096; i += np) {                 // combined biases
        ws[OFF_BIAS + i]          = bih1[i] + bhh1[i];
        ws[OFF_BIAS + 4096 + i]   = bih2[i] + bhh2[i];
        ws[OFF_BIAS + 8192 + i]   = bih3[i] + bhh3[i];
    }
    for (int i = tid; i < FP_; i += np) ws[OFF_BD + i] = (i < F_) ? bd[i] : 0.0f;
    for (int i = tid; i < SZ_HT; i += np) ws[OFF_HT + i] = 0.0f;   // h = 0
    for (int i = tid; i < SZ_C;  i += np) ws[OFF_C  + i] = 0.0f;   // c = 0
    for (int i = tid; i < FP_ * B_; i += np) {             // x_0 transposed (zero padded)
        int k = i >> 5, m = i & 31;
        ws[OFF_XT + htix(k, m)] = (k < F_) ? iseq[(size_t)m * T0_ * F_ + k] : 0.0f;
    }
    int Stot = T0_ + Gp[0];                                // original_seq copy
    size_t obase = (size_t)B_ * Stot * F_;
    for (int i = tid; i < B_ * T0_ * F_; i += np) out[obase + i] = iseq[i];
}

// ---------------- persistent 3-layer LSTM + decoder ----------------
__global__ void __launch_bounds__(256)
acLSTM_persist(const float* __restrict__ iseq, const int* __restrict__ Gp,
               const float* __restrict__ Whh1,
               const float* __restrict__ Wih2, const float* __restrict__ Whh2,
               const float* __restrict__ Wih3, const float* __restrict__ Whh3,
               float* __restrict__ out, float* __restrict__ ws) {
    const int Stot = T0_ + Gp[0];
    unsigned* bar = (unsigned*)(ws + OFF_BAR);
    const unsigned nb = gridDim.x;

    const int wave = threadIdx.x >> 5;
    const int lane = threadIdx.x & 31;
    const int nl = lane & 15;
    const int hi = lane >> 4;
    const int u = blockIdx.x * 8 + wave;       // 0..383: 3 layers x 64 N-tiles x 2 M-tiles

    float* XTp = ws + OFF_XT;

    // per-wave constants (wave-uniform)
    const int layer = u >> 7;
    const int r = u & 127;
    const int m0p1 = (r & 1) << 4;
    const int j0 = (r >> 1) << 4;

    for (int t = 0; t < Stot; ++t) {
        const int rp = t & 1, wp = rp ^ 1;

        // ---- phase 1: gates + cell update for all 3 layers (independent, old-h semantics) ----
        {
            const float* HTb = ws + OFF_HT;
            const float* bC = ws + OFF_BIAS + layer * 4096;
            const float b_i = bC[0 * H_ + j0 + nl];
            const float b_f = bC[1 * H_ + j0 + nl];
            const float b_g = bC[2 * H_ + j0 + nl];
            const float b_o = bC[3 * H_ + j0 + nl];
            v8f ai, af, ag, ao;
#pragma unroll
            for (int e = 0; e < 8; ++e) { ai[e] = b_i; af[e] = b_f; ag[e] = b_g; ao[e] = b_o; }

            if (layer == 0) {
                mm4<FP_, FP_>(XTp,                      ws + OFF_WIH1P, m0p1, j0, nl, hi, ai, af, ag, ao);
                mm4<H_,  H_ >(HTb + (0 * 2 + rp) * 32768, Whh1,         m0p1, j0, nl, hi, ai, af, ag, ao);
            } else if (layer == 1) {
                mm4<H_,  H_ >(HTb + (0 * 2 + rp) * 32768, Wih2,         m0p1, j0, nl, hi, ai, af, ag, ao);
                mm4<H_,  H_ >(HTb + (1 * 2 + rp) * 32768, Whh2,         m0p1, j0, nl, hi, ai, af, ag, ao);
            } else {
                mm4<H_,  H_ >(HTb + (1 * 2 + rp) * 32768, Wih3,         m0p1, j0, nl, hi, ai, af, ag, ao);
                mm4<H_,  H_ >(HTb + (2 * 2 + rp) * 32768, Whh3,         m0p1, j0, nl, hi, ai, af, ag, ao);
            }

            float* cl = ws + OFF_C + layer * 32768;
            float* hw = ws + OFF_HT + (layer * 2 + wp) * 32768;
            const int j = j0 + nl;
#pragma unroll
            for (int e = 0; e < 8; ++e) {
                int m = m0p1 + e + (hi << 3);
                float ig = sigm_(ai[e]);
                float fg = sigm_(af[e]);
                float gg = tanh_(ag[e]);
                float og = sigm_(ao[e]);
                float co = cl[m * H_ + j];
                float cn = fg * co + ig * gg;
                cl[m * H_ + j] = cn;
                hw[htix(j, m)] = og * tanh_(cn);
            }
        }
        gbar(bar, nb);

        // ---- phase 2: decoder out_t = nh2 @ Wd^T + bd ; prep next x ----
        if (u < 22) {                                  // 11 N-tiles x 2 M-tiles
            const int m0 = (u & 1) << 4;
            const int n0 = (u >> 1) << 4;
            const float* S = ws + OFF_HT + (2 * 2 + wp) * 32768;   // new h2
            const float* W = ws + OFF_WDP;
            const float bv = ws[OFF_BD + n0 + nl];
            v8f acc;
#pragma unroll
            for (int e = 0; e < 8; ++e) acc[e] = bv;
            const float* sp = S + hi * 64 + (m0 + nl) * 2;
            const float* wr = W + (size_t)(n0 + nl) * H_ + 2 * hi;
#pragma unroll 4
            for (int kk = 0; kk < H_; kk += 4) {
                v2f a = *(const v2f*)sp; sp += 128;
                v2f b = *(const v2f*)(wr + kk);
                acc = wmma4(a, b, acc);
            }
            const int n = n0 + nl;
            if (n < F_) {
                const bool feed = (t >= T0_ - 1) && (t < Stot - 1);  // autoregressive feedback
                const size_t rowlen = (size_t)Stot * F_;
#pragma unroll
                for (int e = 0; e < 8; ++e) {
                    int m = m0 + e + (hi << 3);
                    out[(size_t)m * rowlen + (size_t)t * F_ + n] = acc[e];
                    if (feed) XTp[htix(n, m)] = acc[e];
                }
            }
        } else if (u >= 24 && u < 32 && (t + 1) < T0_) {
            // conditioned phase: transpose next seed frame into XT
            const int w2 = u - 24;
            for (int idx = w2 * 32 + lane; idx < F_ * B_; idx += 256) {
                int m = idx / F_;
                int k = idx - m * F_;
                XTp[htix(k, m)] = iseq[((size_t)m * T0_ + (t + 1)) * F_ + k];
            }
        }
        gbar(bar, nb);
    }
}

extern "C" void kernel_launch(void* const* d_in, const int* in_sizes, int n_in,
                              void* d_out, int out_size, void* d_ws, size_t ws_size,
                              hipStream_t stream) {
    (void)in_sizes; (void)n_in; (void)out_size; (void)ws_size;
    const float* iseq = (const float*)d_in[0];
    const int*   Gp   = (const int*)d_in[1];
    const float* Wih1 = (const float*)d_in[2];
    const float* Whh1 = (const float*)d_in[3];
    const float* bih1 = (const float*)d_in[4];
    const float* bhh1 = (const float*)d_in[5];
    const float* Wih2 = (const float*)d_in[6];
    const float* Whh2 = (const float*)d_in[7];
    const float* bih2 = (const float*)d_in[8];
    const float* bhh2 = (const float*)d_in[9];
    const float* Wih3 = (const float*)d_in[10];
    const float* Whh3 = (const float*)d_in[11];
    const float* bih3 = (const float*)d_in[12];
    const float* bhh3 = (const float*)d_in[13];
    const float* Wd   = (const float*)d_in[14];
    const float* bd   = (const float*)d_in[15];
    float* out = (float*)d_out;
    float* ws  = (float*)d_ws;

    acLSTM_prepack<<<dim3(256), dim3(256), 0, stream>>>(iseq, Gp, Wih1, bih1, bhh1,
                                                        bih2, bhh2, bih3, bhh3, Wd, bd, out, ws);
    acLSTM_persist<<<dim3(48), dim3(256), 0, stream>>>(iseq, Gp, Whh1, Wih2, Whh2,
                                                       Wih3, Whh3, out, ws);
}